// dict_Encoding_50646254355197
// MI455X (gfx1250) — compile-verified
//
#include <hip/hip_runtime.h>

typedef __attribute__((ext_vector_type(2))) float v2f;
typedef __attribute__((ext_vector_type(8))) float v8f;

#define D_DIM 512
#define K_DIM 32
#define HW    4096
#define POSB  128           // positions per block
#define NTHR  256

// LDS layout (dword offsets), all fragment-major so every fragment read is one ds_load_b64:
//  F1 : [128 d-steps][2 ktiles][32 lanes][2]  = 16384 floats (GEMM1 B-fragments of C)
//  F2 : [32 dtiles][8 ksteps][32 lanes][2]    = 16384 floats (GEMM2 A-fragments of C)
//  FA : per wave [8 ksteps][32 lanes][2]      =   512 floats (GEMM2 B-fragments of softmax A)
#define F1_OFF 0
#define F2_OFF 16384
#define FA_OFF 32768
#define LDS_FLOATS (FA_OFF + 8 * 512)          // 36864 floats = 144 KB

// ds_swizzle group-of-32 XOR butterflies (and_mask=0x1F, or_mask=0, xor in [14:10]).
// XOR masks 1/2/4/8 stay within each 16-lane half — the D-fragment row domain.
#define RED_MAX_STEP(v, imm) \
    v = fmaxf(v, __int_as_float(__builtin_amdgcn_ds_swizzle(__float_as_int(v), imm)))
#define RED_SUM_STEP(v, imm) \
    v = v + __int_as_float(__builtin_amdgcn_ds_swizzle(__float_as_int(v), imm))

__global__ __launch_bounds__(NTHR)
void dict_encoding_wmma_f32(const float* __restrict__ X,
                            const float* __restrict__ CW,
                            const float* __restrict__ gamma,
                            float* __restrict__ out)
{
    extern __shared__ float lds[];
    const int tid  = threadIdx.x;
    const int wave = tid >> 5;
    const int lane = tid & 31;
    const int lh   = lane >> 4;                        // half-wave index
    const int lm   = lane & 15;                        // lane-in-half

    v2f* F1v = (v2f*)(lds + F1_OFF);
    v2f* F2v = (v2f*)(lds + F2_OFF);
    float* FA = lds + FA_OFF + wave * 512;
    const v2f* FAv = (const v2f*)FA;

    const int b   = blockIdx.x >> 5;                   // 32 tiles per batch
    const int til = blockIdx.x & 31;
    const int hw0 = til * POSB + wave * 16;            // this wave's 16 positions
    const float g = gamma[0];

    // ---- stage codewords into fragment-major LDS tables ----
    // F1: GEMM1 B-frag  B(k=v+2lh, n=lm) of column tile t at d-step s
    for (int e = tid; e < 8192; e += NTHR) {
        const int s  = e >> 6;
        const int t  = (e >> 5) & 1;
        const int ln = e & 31;
        const int row = 4 * s + 2 * (ln >> 4);
        const int col = 16 * t + (ln & 15);
        v2f w;
        w.x = CW[row * K_DIM + col];
        w.y = CW[(row + 1) * K_DIM + col];
        F1v[e] = w;
    }
    // F2: GEMM2 A-frag  A'(m=lm -> d=d0+lm, k=4ks+2lh+v)
    for (int e = tid; e < 8192; e += NTHR) {
        const int dt = e >> 8;
        const int ks = (e >> 5) & 7;
        const int ln = e & 31;
        const int row = dt * 16 + (ln & 15);
        const int col = 4 * ks + 2 * (ln >> 4);
        v2f w;
        w.x = CW[row * K_DIM + col];
        w.y = CW[row * K_DIM + col + 1];
        F2v[e] = w;
    }
    __syncthreads();

    const float* Xb = X + (size_t)b * D_DIM * HW;

    // ---- GEMM1: logits[16 pos x 32 k] = X_tile[16 x 512] @ C[512 x 32] ----
    v8f s0 = {};   // k = 0..15
    v8f s1 = {};   // k = 16..31
#pragma unroll 8
    for (int d0 = 0; d0 < D_DIM; d0 += 4) {
        const int s  = d0 >> 2;
        const int ka = d0 + 2 * lh;
        v2f a;
        a.x = Xb[(size_t)(ka    ) * HW + hw0 + lm];    // A(m=lm, k=2lh)
        a.y = Xb[(size_t)(ka + 1) * HW + hw0 + lm];    // A(m=lm, k=2lh+1)
        const v2f b0 = F1v[(s * 2 + 0) * 32 + lane];   // cols 0..15
        const v2f b1 = F1v[(s * 2 + 1) * 32 + lane];   // cols 16..31
        s0 = __builtin_amdgcn_wmma_f32_16x16x4_f32(false, a, false, b0, (short)0, s0, false, false);
        s1 = __builtin_amdgcn_wmma_f32_16x16x4_f32(false, a, false, b1, (short)0, s1, false, false);
    }

    // ---- softmax over K=32 per position; write straight into fragment order ----
    // value (pos,k) -> FA[(k>>2)][((k>>1)&1)*16 + pos][k&1]
    const int fa0 = ((lm >> 2) * 32 + ((lm >> 1) & 1) * 16) * 2 + (lm & 1);
#pragma unroll
    for (int j = 0; j < 8; ++j) {
        float a0 = s0[j], a1 = s1[j];                  // (pos = j + 8*lh, k = lm / lm+16)
        float mx = fmaxf(a0, a1);
        RED_MAX_STEP(mx, 0x041F);
        RED_MAX_STEP(mx, 0x081F);
        RED_MAX_STEP(mx, 0x101F);
        RED_MAX_STEP(mx, 0x201F);
        float e0 = __expf(a0 - mx);
        float e1 = __expf(a1 - mx);
        float sm = e0 + e1;
        RED_SUM_STEP(sm, 0x041F);
        RED_SUM_STEP(sm, 0x081F);
        RED_SUM_STEP(sm, 0x101F);
        RED_SUM_STEP(sm, 0x201F);
        const float inv = 1.0f / sm;
        const int pos = j + 8 * lh;
        FA[fa0 + pos * 2]       = e0 * inv;            // k = lm
        FA[fa0 + pos * 2 + 256] = e1 * inv;            // k = lm + 16 (+4 ksteps * 64)
    }

    // ---- GEMM2 (transposed): Yt[16 d x 16 pos] = C_tile[16 x 32] @ A^T[32 x 16] ----
    // M = d rows -> D-frag rows hit consecutive hw addresses: coalesced stores.
    for (int dt = 0; dt < D_DIM / 16; ++dt) {
        const int d0 = dt * 16;

        // pull the residual-read X lines toward the WGP while WMMAs run
        __builtin_prefetch(&X[((size_t)b * D_DIM + d0 + 8 * lh) * HW + hw0 + lm], 0, 3);

        v8f y = {};
#pragma unroll
        for (int ks = 0; ks < 8; ++ks) {
            const v2f ca = F2v[(dt * 8 + ks) * 32 + lane];   // A'(m=d, k)
            const v2f ab = FAv[ks * 32 + lane];              // B'(k, n=pos)
            y = __builtin_amdgcn_wmma_f32_16x16x4_f32(false, ca, false, ab, (short)0, y, false, false);
        }
#pragma unroll
        for (int j = 0; j < 8; ++j) {
            const int d = d0 + j + 8 * lh;
            const size_t idx = ((size_t)b * D_DIM + d) * HW + hw0 + lm;
            out[idx] = X[idx] + g * y[j];              // residual; X re-read is L2-hot
        }
    }
}

extern "C" void kernel_launch(void* const* d_in, const int* in_sizes, int n_in,
                              void* d_out, int out_size, void* d_ws, size_t ws_size,
                              hipStream_t stream) {
    const float* X     = (const float*)d_in[0];
    const float* CW    = (const float*)d_in[1];
    const float* gamma = (const float*)d_in[2];
    float* out = (float*)d_out;

    const size_t shmem = (size_t)LDS_FLOATS * sizeof(float);   // 147,456 B
    dim3 grid(16 * (HW / POSB));   // 16 batches * 32 tiles = 512 blocks
    dim3 block(NTHR);
    dict_encoding_wmma_f32<<<grid, block, shmem, stream>>>(X, CW, gamma, out);
}